// DCNNv2_41051297415545
// MI455X (gfx1250) — compile-verified
//
#include <hip/hip_runtime.h>
#include <math.h>

typedef __attribute__((ext_vector_type(2))) float v2f;
typedef __attribute__((ext_vector_type(8))) float v8f;

#define DD   128
#define LDX  132   // LDS row stride (floats): 132 % 64 = 4 -> conflict-free across 16 rows
#define LDX2 260   // 256-wide tile stride, same property

__device__ __forceinline__ v8f wmma_f32_k4(v2f a, v2f b, v8f c) {
    // 8 args: (neg_a, A, neg_b, B, c_mod, C, reuse_a, reuse_b)
    return __builtin_amdgcn_wmma_f32_16x16x4_f32(false, a, false, b, (short)0, c, false, false);
}

// ---------------------------------------------------------------------------
// Stage 1: per node n (one block): gather emb (16x128) and nb_sum (16x128),
// S = relu(X*W^T + Y*M^T), r = column-sum over 16 rows, h = softmax(r)
// ---------------------------------------------------------------------------
__global__ __launch_bounds__(128)
void node_stage1(const int* __restrict__ nid, const int* __restrict__ nbr,
                 const float* __restrict__ E, const float* __restrict__ W,
                 const float* __restrict__ M, float* __restrict__ hout) {
    __shared__ float Xs[16 * LDX];
    __shared__ float Ys[16 * LDX];
    __shared__ float rbuf[DD];
    __shared__ float redm[4], reds[4];
    __shared__ int   sNid[16];
    __shared__ int   sNbr[128];

    const int n    = blockIdx.x;
    const int t    = threadIdx.x;
    const int lane = t & 31, wave = t >> 5;
    const int r16  = lane & 15, kh = lane >> 4;

    if (t < 16) sNid[t] = nid[n * 16 + t];
    sNbr[t] = nbr[n * 128 + t];          // 16 k-slots * 8 neighbors = 128 ids
    __syncthreads();

    // Gather (float4): wave w owns rows w, w+4, w+8, w+12; lane covers cols 4*lane..4*lane+3
    {
        const int c4 = lane * 4;
#pragma unroll
        for (int kq = 0; kq < 4; ++kq) {
            const int k = wave + kq * 4;
            *(float4*)&Xs[k * LDX + c4] =
                *(const float4*)&E[(size_t)sNid[k] * DD + c4];
            float4 acc = make_float4(0.f, 0.f, 0.f, 0.f);
#pragma unroll
            for (int j = 0; j < 8; ++j) {
                const float4 v = *(const float4*)&E[(size_t)sNbr[k * 8 + j] * DD + c4];
                acc.x += v.x; acc.y += v.y; acc.z += v.z; acc.w += v.w;
            }
            *(float4*)&Ys[k * LDX + c4] = acc;
        }
    }
    __syncthreads();

    v8f acc0 = {}; v8f acc1 = {};
    const int c0 = wave * 16 + r16;
    const int c1 = (wave + 4) * 16 + r16;
    for (int kk = 0; kk < 32; ++kk) {
        const int kb = kk * 4 + kh * 2;
        v2f a  = *(const v2f*)&Xs[r16 * LDX + kb];
        v2f b0 = *(const v2f*)&W[c0 * DD + kb];
        v2f b1 = *(const v2f*)&W[c1 * DD + kb];
        acc0 = wmma_f32_k4(a, b0, acc0);
        acc1 = wmma_f32_k4(a, b1, acc1);
    }
    for (int kk = 0; kk < 32; ++kk) {
        const int kb = kk * 4 + kh * 2;
        v2f a  = *(const v2f*)&Ys[r16 * LDX + kb];
        v2f b0 = *(const v2f*)&M[c0 * DD + kb];
        v2f b1 = *(const v2f*)&M[c1 * DD + kb];
        acc0 = wmma_f32_k4(a, b0, acc0);
        acc1 = wmma_f32_k4(a, b1, acc1);
    }

    // relu + 16-row column reduction (lane l holds rows v+8*kh of column l&15)
    float p0 = 0.f, p1 = 0.f;
#pragma unroll
    for (int v = 0; v < 8; ++v) {
        p0 += fmaxf(acc0[v], 0.f);
        p1 += fmaxf(acc1[v], 0.f);
    }
    p0 += __shfl_xor(p0, 16, 32);
    p1 += __shfl_xor(p1, 16, 32);
    if (lane < 16) { rbuf[c0] = p0; rbuf[c1] = p1; }
    __syncthreads();

    // softmax over 128 columns
    float x = rbuf[t];
    float m = x;
    for (int off = 16; off > 0; off >>= 1) m = fmaxf(m, __shfl_xor(m, off, 32));
    if (lane == 0) redm[wave] = m;
    __syncthreads();
    m = fmaxf(fmaxf(redm[0], redm[1]), fmaxf(redm[2], redm[3]));
    float e = __expf(x - m);
    float s = e;
    for (int off = 16; off > 0; off >>= 1) s += __shfl_xor(s, off, 32);
    if (lane == 0) reds[wave] = s;
    __syncthreads();
    s = reds[0] + reds[1] + reds[2] + reds[3];
    hout[(size_t)n * DD + t] = e / s;
}

// ---------------------------------------------------------------------------
// Stage 2: 16 nodes per block: X=h rows, Y=ext_sum rows,
// e_all = row-softmax(relu(X*U^T + Y*V^T))
// ---------------------------------------------------------------------------
__global__ __launch_bounds__(128)
void node_stage2(const int* __restrict__ ext, const float* __restrict__ hin,
                 const float* __restrict__ U, const float* __restrict__ V,
                 float* __restrict__ eout) {
    __shared__ float Xs[16 * LDX];
    __shared__ float Ys[16 * LDX];
    __shared__ float Ss[16 * LDX];
    __shared__ int   sExt[256];

    const int nb   = blockIdx.x * 16;
    const int t    = threadIdx.x;
    const int lane = t & 31, wave = t >> 5;
    const int r16  = lane & 15, kh = lane >> 4;

    sExt[t]       = ext[nb * 16 + t];
    sExt[t + 128] = ext[nb * 16 + t + 128];
    __syncthreads();

    // Gather (float4): wave w owns rows w, w+4, w+8, w+12
    {
        const int c4 = lane * 4;
#pragma unroll
        for (int kq = 0; kq < 4; ++kq) {
            const int k = wave + kq * 4;
            *(float4*)&Xs[k * LDX + c4] =
                *(const float4*)&hin[(size_t)(nb + k) * DD + c4];
            float4 acc = make_float4(0.f, 0.f, 0.f, 0.f);
#pragma unroll
            for (int j = 0; j < 16; ++j) {
                const float4 v = *(const float4*)&hin[(size_t)sExt[k * 16 + j] * DD + c4];
                acc.x += v.x; acc.y += v.y; acc.z += v.z; acc.w += v.w;
            }
            *(float4*)&Ys[k * LDX + c4] = acc;
        }
    }
    __syncthreads();

    v8f acc0 = {}; v8f acc1 = {};
    const int c0 = wave * 16 + r16;
    const int c1 = (wave + 4) * 16 + r16;
    for (int kk = 0; kk < 32; ++kk) {
        const int kb = kk * 4 + kh * 2;
        v2f a  = *(const v2f*)&Xs[r16 * LDX + kb];
        v2f b0 = *(const v2f*)&U[c0 * DD + kb];
        v2f b1 = *(const v2f*)&U[c1 * DD + kb];
        acc0 = wmma_f32_k4(a, b0, acc0);
        acc1 = wmma_f32_k4(a, b1, acc1);
    }
    for (int kk = 0; kk < 32; ++kk) {
        const int kb = kk * 4 + kh * 2;
        v2f a  = *(const v2f*)&Ys[r16 * LDX + kb];
        v2f b0 = *(const v2f*)&V[c0 * DD + kb];
        v2f b1 = *(const v2f*)&V[c1 * DD + kb];
        acc0 = wmma_f32_k4(a, b0, acc0);
        acc1 = wmma_f32_k4(a, b1, acc1);
    }
#pragma unroll
    for (int v = 0; v < 8; ++v) {
        const int row = v + 8 * kh;
        Ss[row * LDX + c0] = fmaxf(acc0[v], 0.f);
        Ss[row * LDX + c1] = fmaxf(acc1[v], 0.f);
    }
    __syncthreads();

    // per-row softmax: 8 threads per row
    const int row = t >> 3, g = t & 7;
    float m = -1e30f;
    for (int c = g; c < DD; c += 8) m = fmaxf(m, Ss[row * LDX + c]);
    for (int off = 4; off > 0; off >>= 1) m = fmaxf(m, __shfl_xor(m, off, 32));
    float s = 0.f;
    for (int c = g; c < DD; c += 8) s += __expf(Ss[row * LDX + c] - m);
    for (int off = 4; off > 0; off >>= 1) s += __shfl_xor(s, off, 32);
    const float inv = 1.f / s;
    for (int c = g; c < DD; c += 8)
        eout[(size_t)(nb + row) * DD + c] = __expf(Ss[row * LDX + c] - m) * inv;
}

// ---------------------------------------------------------------------------
// Head: 16 batch rows per block: x = [e_all[a] | e_all[b]] (16x256),
// hid = leaky_relu(x*W1^T + b1), logits = hid*W2^T + b2, softmax(2)
// ---------------------------------------------------------------------------
__global__ __launch_bounds__(128)
void head_stage(const int* __restrict__ batch, const float* __restrict__ eall,
                const float* __restrict__ W1, const float* __restrict__ b1,
                const float* __restrict__ W2, const float* __restrict__ b2,
                float* __restrict__ out) {
    __shared__ float Xs[16 * LDX2];
    __shared__ float Hs[16 * LDX];
    __shared__ int   sA[16], sB[16];

    const int rb   = blockIdx.x * 16;
    const int t    = threadIdx.x;
    const int lane = t & 31, wave = t >> 5;
    const int r16  = lane & 15, kh = lane >> 4;

    if (t < 16)      sA[t]      = batch[(rb + t) * 2 + 0];
    else if (t < 32) sB[t - 16] = batch[(rb + t - 16) * 2 + 1];
    __syncthreads();

    // Gather (float4): wave w owns rows w, w+4, w+8, w+12
    {
        const int c4 = lane * 4;
#pragma unroll
        for (int iq = 0; iq < 4; ++iq) {
            const int i = wave + iq * 4;
            *(float4*)&Xs[i * LDX2 + c4] =
                *(const float4*)&eall[(size_t)sA[i] * DD + c4];
            *(float4*)&Xs[i * LDX2 + DD + c4] =
                *(const float4*)&eall[(size_t)sB[i] * DD + c4];
        }
    }
    __syncthreads();

    v8f acc0 = {}; v8f acc1 = {};
    const int c0 = wave * 16 + r16;
    const int c1 = (wave + 4) * 16 + r16;
    for (int kk = 0; kk < 64; ++kk) {   // K = 256
        const int kb = kk * 4 + kh * 2;
        v2f a  = *(const v2f*)&Xs[r16 * LDX2 + kb];
        v2f b0 = *(const v2f*)&W1[c0 * 256 + kb];
        v2f b1 = *(const v2f*)&W1[c1 * 256 + kb];
        acc0 = wmma_f32_k4(a, b0, acc0);
        acc1 = wmma_f32_k4(a, b1, acc1);
    }
    const float bb0 = b1[c0], bb1 = b1[c1];
#pragma unroll
    for (int v = 0; v < 8; ++v) {
        const int row = v + 8 * kh;
        float x0 = acc0[v] + bb0;
        float x1 = acc1[v] + bb1;
        Hs[row * LDX + c0] = (x0 > 0.f) ? x0 : 0.01f * x0;  // leaky_relu(0.01)
        Hs[row * LDX + c1] = (x1 > 0.f) ? x1 : 0.01f * x1;
    }
    __syncthreads();

    const int row = t >> 3, g = t & 7;
    float l0 = 0.f, l1 = 0.f;
    for (int c = g; c < DD; c += 8) {
        const float hv = Hs[row * LDX + c];
        l0 += hv * W2[c];
        l1 += hv * W2[DD + c];
    }
    for (int off = 4; off > 0; off >>= 1) {
        l0 += __shfl_xor(l0, off, 32);
        l1 += __shfl_xor(l1, off, 32);
    }
    if (g == 0) {
        l0 += b2[0]; l1 += b2[1];
        const float m  = fmaxf(l0, l1);
        const float e0 = __expf(l0 - m), e1 = __expf(l1 - m);
        const float inv = 1.f / (e0 + e1);
        out[(size_t)(rb + row) * 2 + 0] = e0 * inv;
        out[(size_t)(rb + row) * 2 + 1] = e1 * inv;
    }
}

// ---------------------------------------------------------------------------
extern "C" void kernel_launch(void* const* d_in, const int* in_sizes, int n_in,
                              void* d_out, int out_size, void* d_ws, size_t ws_size,
                              hipStream_t stream) {
    (void)in_sizes; (void)n_in; (void)out_size; (void)ws_size;
    const int*   batch = (const int*)d_in[0];
    const int*   nid   = (const int*)d_in[1];
    const int*   nbr   = (const int*)d_in[2];
    const int*   ext   = (const int*)d_in[3];
    const float* E     = (const float*)d_in[4];
    const float* W     = (const float*)d_in[5];
    const float* M     = (const float*)d_in[6];
    const float* U     = (const float*)d_in[7];
    const float* V     = (const float*)d_in[8];
    const float* W1    = (const float*)d_in[9];
    const float* b1    = (const float*)d_in[10];
    const float* W2    = (const float*)d_in[11];
    const float* b2    = (const float*)d_in[12];

    float* hbuf = (float*)d_ws;                       // 10000*128 f32
    float* ebuf = hbuf + (size_t)10000 * 128;         // 10000*128 f32

    node_stage1<<<10000, 128, 0, stream>>>(nid, nbr, E, W, M, hbuf);
    node_stage2<<<625,   128, 0, stream>>>(ext, hbuf, U, V, ebuf);
    head_stage <<<128,   128, 0, stream>>>(batch, ebuf, W1, b1, W2, b2, (float*)d_out);
}